// I2I_OTFeaturesFusion_8340826489182
// MI455X (gfx1250) — compile-verified
//
#include <hip/hip_runtime.h>
#include <math.h>

// ---------------------------------------------------------------------------
// I2I_OTFeaturesFusion for MI455X (gfx1250, wave32)
//   B=2, C=64, M=D*H*W=2048, EPS=0.1, 50 Sinkhorn iterations.
// Strategy: immutable log-kernel + dual potentials (u,v); lk0 and lk0^T kept
// resident in the 192MB L2; both GEMMs use V_WMMA_F32_16X16X4_F32 (exact f32).
// ---------------------------------------------------------------------------

#define EPS_F 0.1f
#define N_ITERS 50
#define BATCH 2
#define CCH 64            // channels C
#define MM 2048           // M = D*H*W

typedef __attribute__((ext_vector_type(2))) float v2f;
typedef __attribute__((ext_vector_type(8))) float v8f;

__device__ __forceinline__ v8f wmma_f32_4(v2f a, v2f b, v8f c) {
    // V_WMMA_F32_16X16X4_F32 : D = A(16x4) * B(4x16) + C(16x16), all fp32
    return __builtin_amdgcn_wmma_f32_16x16x4_f32(
        false, a, false, b, (short)0, c, false, false);
}

// ---------------------------------------------------------------------------
// Kernel 1: per-point squared norms, seq2 transpose ([b][m][c]), zero u/v.
// feat layout in memory: [B][C][M] (M contiguous).
// ---------------------------------------------------------------------------
__global__ void prep_kernel(const float* __restrict__ feat1,
                            const float* __restrict__ feat2,
                            float* __restrict__ sq1, float* __restrict__ sq2,
                            float* __restrict__ seq2t,
                            float* __restrict__ u, float* __restrict__ v) {
    int g = blockIdx.x * blockDim.x + threadIdx.x;       // 0 .. B*M-1
    int b = g >> 11, m = g & (MM - 1);
    const float* f1 = feat1 + (size_t)b * CCH * MM;
    const float* f2 = feat2 + (size_t)b * CCH * MM;
    float* s2t = seq2t + (size_t)b * MM * CCH + (size_t)m * CCH;
    float a1 = 0.f, a2 = 0.f;
    for (int c = 0; c < CCH; ++c) {
        float x = f1[(size_t)c * MM + m];
        float y = f2[(size_t)c * MM + m];
        a1 = fmaf(x, x, a1);
        a2 = fmaf(y, y, a2);
        s2t[c] = y;
    }
    sq1[g] = a1;
    sq2[g] = a2;
    u[g] = 0.0f;
    v[g] = 0.0f;
}

// ---------------------------------------------------------------------------
// Kernel 2: lk0 = -max(sq1[m]+sq2[n]-2*cross[m,n],0)/EPS via fp32 WMMA.
// One wave per 16x16 tile; 16 K-chunks of 4 (C=64). Writes lk0 and lk0^T.
// A fragment layout (16x4 f32): lane L: M = L%16, K = 2*(L/16)+{0,1}.
// B fragment (4x16, column-per-lane): lane L: N = L%16, K = 2*(L/16)+{0,1}.
// C/D (16x16 f32): vgpr r, lane L -> (M = r + 8*(L/16), N = L%16).
// ---------------------------------------------------------------------------
__global__ void cost_wmma_kernel(const float* __restrict__ feat1,
                                 const float* __restrict__ feat2,
                                 const float* __restrict__ sq1,
                                 const float* __restrict__ sq2,
                                 float* __restrict__ lk0,
                                 float* __restrict__ lk0T) {
    int wave = blockIdx.x * (blockDim.x >> 5) + (threadIdx.x >> 5);
    int b = wave >> 14;                 // 128*128 tiles per batch
    int t = wave & 16383;
    int mt = t >> 7, nt = t & 127;
    int lane = threadIdx.x & 31;
    int row = lane & 15, half = lane >> 4;
    int m0 = mt << 4, n0 = nt << 4;

    const float* f1 = feat1 + (size_t)b * CCH * MM;
    const float* f2 = feat2 + (size_t)b * CCH * MM;

    v8f acc = {};
    for (int k0 = 0; k0 < CCH; k0 += 4) {
        int ka = k0 + 2 * half;
        v2f a, bb;
        a.x  = f1[(size_t)ka * MM + m0 + row];
        a.y  = f1[(size_t)(ka + 1) * MM + m0 + row];
        bb.x = f2[(size_t)ka * MM + n0 + row];
        bb.y = f2[(size_t)(ka + 1) * MM + n0 + row];
        acc = wmma_f32_4(a, bb, acc);
    }

    const float* s1 = sq1 + b * MM;
    const float* s2 = sq2 + b * MM;
    float* l0  = lk0  + (size_t)b * MM * MM;
    float* l0t = lk0T + (size_t)b * MM * MM;
    int n = n0 + row;
    float s2n = s2[n];
    #pragma unroll
    for (int r = 0; r < 8; ++r) {
        int m = m0 + r + 8 * half;
        float cval = fmaxf(s1[m] + s2n - 2.0f * acc[r], 0.0f);
        float lk = cval * (-1.0f / EPS_F);
        l0[(size_t)m * MM + n]  = lk;   // coalesced (row-major)
        l0t[(size_t)n * MM + m] = lk;   // 16x64B footprint, L2 write-combined
    }
}

// ---------------------------------------------------------------------------
// Kernel 3: one Sinkhorn half-step. pout[r] = -lse_n( lk[r][n] + pin[n] ).
// One 256-thread block per row; 8 coalesced elements per thread held in regs.
// Called with (lk0, v -> u) then (lk0T, u -> v); data stays hot in L2.
// ---------------------------------------------------------------------------
__global__ void sinkhorn_half_kernel(const float* __restrict__ lk,
                                     const float* __restrict__ pin,
                                     float* __restrict__ pout) {
    __shared__ float red[256];
    int rowg = blockIdx.x;                 // 0 .. B*M-1
    int b = rowg >> 11, m = rowg & (MM - 1);
    const float* base = lk + (size_t)b * MM * MM + (size_t)m * MM;
    const float* pv = pin + b * MM;
    int t = threadIdx.x;

    float x[8];
    float mx = -3.402823466e38f;
    #pragma unroll
    for (int k = 0; k < 8; ++k) {
        int n = t + (k << 8);
        x[k] = base[n] + pv[n];
        mx = fmaxf(mx, x[k]);
    }
    red[t] = mx;
    __syncthreads();
    for (int s = 128; s > 0; s >>= 1) {
        if (t < s) red[t] = fmaxf(red[t], red[t + s]);
        __syncthreads();
    }
    float gmax = red[0];
    __syncthreads();
    float sm = 0.f;
    #pragma unroll
    for (int k = 0; k < 8; ++k) sm += __expf(x[k] - gmax);
    red[t] = sm;
    __syncthreads();
    for (int s = 128; s > 0; s >>= 1) {
        if (t < s) red[t] += red[t + s];
        __syncthreads();
    }
    if (t == 0) pout[rowg] = -(gmax + __logf(red[0]));
}

// ---------------------------------------------------------------------------
// Kernel 4: fused_ot_t[b][m][c] = sum_n exp(lk0[m][n]+u[m]+v[n]) * seq2t[n][c]
// via fp32 WMMA, K = n (2048, chunks of 4). One wave per 16-row M strip
// computes all 4 C-tiles (A fragment + its 2 exps reused 4x).
// A read from lk0T so lanes 0-15 hit consecutive addresses.
// ---------------------------------------------------------------------------
__global__ void pgemm_wmma_kernel(const float* __restrict__ lk0T,
                                  const float* __restrict__ u,
                                  const float* __restrict__ v,
                                  const float* __restrict__ seq2t,
                                  float* __restrict__ fot) {
    int wave = blockIdx.x * (blockDim.x >> 5) + (threadIdx.x >> 5);
    int b = wave >> 7;              // 128 m-tiles per batch
    int mt = wave & 127;
    int lane = threadIdx.x & 31;
    int row = lane & 15, half = lane >> 4;
    int m0 = mt << 4;

    const float* lt = lk0T + (size_t)b * MM * MM;     // [n][m]
    const float* ub = u + b * MM;
    const float* vb = v + b * MM;
    const float* s2 = seq2t + (size_t)b * MM * CCH;   // [n][c]

    v8f acc0 = {}, acc1 = {}, acc2 = {}, acc3 = {};
    float um = ub[m0 + row];                          // per-lane constant

    for (int n0 = 0; n0 < MM; n0 += 4) {
        int na = n0 + 2 * half;
        v2f a;
        a.x = __expf(lt[(size_t)na * MM + m0 + row] + vb[na] + um);
        a.y = __expf(lt[(size_t)(na + 1) * MM + m0 + row] + vb[na + 1] + um);
        const float* r0 = s2 + (size_t)na * CCH;
        const float* r1 = r0 + CCH;
        v2f b0, b1, b2, b3;
        b0.x = r0[row];      b0.y = r1[row];
        b1.x = r0[16 + row]; b1.y = r1[16 + row];
        b2.x = r0[32 + row]; b2.y = r1[32 + row];
        b3.x = r0[48 + row]; b3.y = r1[48 + row];
        acc0 = wmma_f32_4(a, b0, acc0);
        acc1 = wmma_f32_4(a, b1, acc1);
        acc2 = wmma_f32_4(a, b2, acc2);
        acc3 = wmma_f32_4(a, b3, acc3);
    }

    float* o = fot + (size_t)b * MM * CCH;            // [m][c]
    #pragma unroll
    for (int r = 0; r < 8; ++r) {
        size_t mo = (size_t)(m0 + r + 8 * half) * CCH;
        o[mo + row]      = acc0[r];
        o[mo + 16 + row] = acc1[r];
        o[mo + 32 + row] = acc2[r];
        o[mo + 48 + row] = acc3[r];
    }
}

// ---------------------------------------------------------------------------
// Kernel 5: gating fusion. Per point m: h = relu(W1 [f1;fot] + b1),
// gate = sigmoid(W2 h + b2), out = gate*f1 + (1-gate)*fot.
// 4 points per 256-thread block, channel-per-thread via LDS.
// ---------------------------------------------------------------------------
__global__ void gating_kernel(const float* __restrict__ feat1,
                              const float* __restrict__ fot_t,
                              const float* __restrict__ w1,
                              const float* __restrict__ b1,
                              const float* __restrict__ w2,
                              const float* __restrict__ b2,
                              float* __restrict__ out) {
    __shared__ float sf1[4][CCH], sfo[4][CCH], sh[4][CCH];
    int sub = threadIdx.x >> 6;         // 0..3
    int o = threadIdx.x & 63;           // channel
    int g = blockIdx.x * 4 + sub;       // 0 .. B*M-1
    int b = g >> 11, m = g & (MM - 1);

    const float* f1 = feat1 + (size_t)b * CCH * MM;
    const float* fo = fot_t + (size_t)b * MM * CCH + (size_t)m * CCH;
    sf1[sub][o] = f1[(size_t)o * MM + m];
    sfo[sub][o] = fo[o];
    __syncthreads();

    float acc = b1[o];
    const float* w1r = w1 + o * 2 * CCH;
    #pragma unroll 4
    for (int c = 0; c < CCH; ++c) acc = fmaf(w1r[c], sf1[sub][c], acc);
    #pragma unroll 4
    for (int c = 0; c < CCH; ++c) acc = fmaf(w1r[CCH + c], sfo[sub][c], acc);
    sh[sub][o] = fmaxf(acc, 0.0f);
    __syncthreads();

    float ga = b2[o];
    const float* w2r = w2 + o * CCH;
    #pragma unroll 4
    for (int c = 0; c < CCH; ++c) ga = fmaf(w2r[c], sh[sub][c], ga);
    float gate = 1.0f / (1.0f + __expf(-ga));
    out[(size_t)b * CCH * MM + (size_t)o * MM + m] =
        gate * sf1[sub][o] + (1.0f - gate) * sfo[sub][o];
}

// ---------------------------------------------------------------------------
extern "C" void kernel_launch(void* const* d_in, const int* in_sizes, int n_in,
                              void* d_out, int out_size, void* d_ws, size_t ws_size,
                              hipStream_t stream) {
    (void)in_sizes; (void)n_in; (void)out_size; (void)ws_size;
    const float* feat1 = (const float*)d_in[0];
    const float* feat2 = (const float*)d_in[1];
    const float* w1    = (const float*)d_in[2];
    const float* b1    = (const float*)d_in[3];
    const float* w2    = (const float*)d_in[4];
    const float* b2    = (const float*)d_in[5];
    float* out = (float*)d_out;

    // Workspace layout (bytes)
    char* ws = (char*)d_ws;
    float* lk0   = (float*)(ws);                          // 33,554,432 B
    float* lk0T  = (float*)(ws + 33554432);               // 33,554,432 B
    float* seq2t = (float*)(ws + 67108864);               //  2,097,152 B
    float* fot   = (float*)(ws + 69206016);               //  2,097,152 B
    float* sq1   = (float*)(ws + 71303168);               //     16,384 B
    float* sq2   = (float*)(ws + 71319552);               //     16,384 B
    float* upot  = (float*)(ws + 71335936);               //     16,384 B
    float* vpot  = (float*)(ws + 71352320);               //     16,384 B
    // total ~71.37 MB

    const int BM = BATCH * MM;                            // 4096

    // 1. norms + seq2 transpose + zero potentials
    prep_kernel<<<BM / 256, 256, 0, stream>>>(feat1, feat2, sq1, sq2, seq2t,
                                              upot, vpot);

    // 2. cost matrix (WMMA): 2*128*128 tiles, 8 waves/block
    cost_wmma_kernel<<<(BATCH * 128 * 128) / 8, 256, 0, stream>>>(
        feat1, feat2, sq1, sq2, lk0, lk0T);

    // 3. 50 Sinkhorn iterations over L2-resident lk0 / lk0T
    for (int it = 0; it < N_ITERS; ++it) {
        sinkhorn_half_kernel<<<BM, 256, 0, stream>>>(lk0,  vpot, upot); // rows
        sinkhorn_half_kernel<<<BM, 256, 0, stream>>>(lk0T, upot, vpot); // cols
    }

    // 4. P @ seq2 with fused exp (WMMA): 256 waves, 8 waves/block
    pgemm_wmma_kernel<<<(BATCH * 128) / 8, 256, 0, stream>>>(
        lk0T, upot, vpot, seq2t, fot);

    // 5. gating fusion -> output [B][C][M]
    gating_kernel<<<BM / 4, 256, 0, stream>>>(feat1, fot, w1, b1, w2, b2, out);
}